// FeCAM_Manager_25185688224305
// MI455X (gfx1250) — compile-verified
//
#include <hip/hip_runtime.h>

#define B_SZ 32
#define C_SZ 256
#define D_SZ 768
#define LDS_STRIDE 772   // 768 + 4 floats pad: avoid 64-bank stride aliasing
#define NT 6             // N-tiles (of 16 cols) per wave; 8 waves * 6 * 16 = 768

typedef __attribute__((ext_vector_type(2))) float v2f;
typedef __attribute__((ext_vector_type(8))) float v8f;

__global__ __launch_bounds__(256, 1)
void fecam_score_kernel(const float* __restrict__ x,      // [B, D]
                        const float* __restrict__ means,  // [C, D]
                        const float* __restrict__ diags,  // [C, D]
                        const float* __restrict__ covinv, // [C, D, D]
                        float* __restrict__ out) {        // [B, C]
    __shared__ float lds_diff[B_SZ * LDS_STRIDE];
    __shared__ float lds_dist[B_SZ];

    const int c = blockIdx.x;
    const int t = threadIdx.x;
    if (t < B_SZ) lds_dist[t] = 0.0f;

    // ---- Phase 1: diff = (x - mean_c) / diag_c  ->  LDS [32 x 768] ----
    for (int k = t; k < D_SZ; k += 256) {
        const float inv = 1.0f / diags[c * D_SZ + k];
        const float mu  = means[c * D_SZ + k] * inv;
#pragma unroll
        for (int m = 0; m < B_SZ; ++m) {
            lds_diff[m * LDS_STRIDE + k] = x[m * D_SZ + k] * inv - mu;
        }
    }
    __syncthreads();

    // ---- Phase 2: temp = diff @ covinv_c via V_WMMA_F32_16X16X4_F32 ----
    const int wave = t >> 5;
    const int lane = t & 31;
    const int nlo  = lane & 15;        // A: row M, B: col N
    const int khi  = (lane >> 4) * 2;  // K sub-offset for A/B fragments

    const float* covc  = covinv + (size_t)c * D_SZ * D_SZ;
    const float* bbase = covc + (size_t)khi * D_SZ + nlo;   // per-lane B base
    const int n0base   = wave * (NT * 16);

    const float* arow0 = &lds_diff[nlo * LDS_STRIDE + khi];         // M 0..15
    const float* arow1 = &lds_diff[(nlo + 16) * LDS_STRIDE + khi];  // M 16..31

    v8f acc0[NT], acc1[NT];
    const v8f vzero = {};
#pragma unroll
    for (int j = 0; j < NT; ++j) { acc0[j] = vzero; acc1[j] = vzero; }

    for (int k0 = 0; k0 < D_SZ; k0 += 4) {
        const v2f a0 = *(const v2f*)(arow0 + k0);
        const v2f a1 = *(const v2f*)(arow1 + k0);
        const float* bk = bbase + (size_t)k0 * D_SZ + n0base;
#pragma unroll
        for (int j = 0; j < NT; ++j) {
            v2f bf;
            bf.x = __builtin_nontemporal_load(bk + j * 16);         // K = khi
            bf.y = __builtin_nontemporal_load(bk + j * 16 + D_SZ);  // K = khi+1
            acc0[j] = __builtin_amdgcn_wmma_f32_16x16x4_f32(
                false, a0, false, bf, (short)0, acc0[j], false, false);
            acc1[j] = __builtin_amdgcn_wmma_f32_16x16x4_f32(
                false, a1, false, bf, (short)0, acc1[j], false, false);
        }
    }

    // ---- Phase 3: dist[m] += sum_e temp[m][e] * diff[m][e] ----
    // acc layout: vgpr r: lanes 0-15 -> M=r, N=nlo ; lanes 16-31 -> M=r+8
    float dsum0[8], dsum1[8];
#pragma unroll
    for (int r = 0; r < 8; ++r) { dsum0[r] = 0.0f; dsum1[r] = 0.0f; }

#pragma unroll
    for (int j = 0; j < NT; ++j) {
        const int e = n0base + j * 16 + nlo;
#pragma unroll
        for (int r = 0; r < 8; ++r) {
            const int m0 = (lane < 16) ? r : (r + 8);
            float p0 = acc0[j][r] * lds_diff[m0 * LDS_STRIDE + e];
            float p1 = acc1[j][r] * lds_diff[(m0 + 16) * LDS_STRIDE + e];
            // reduce across each 16-lane half (masks < 16 stay in-half)
#pragma unroll
            for (int off = 1; off < 16; off <<= 1) {
                p0 += __shfl_xor(p0, off, 32);
                p1 += __shfl_xor(p1, off, 32);
            }
            dsum0[r] += p0;   // row r (lanes 0-15) / r+8 (lanes 16-31)
            dsum1[r] += p1;   // +16
        }
    }
    if (nlo == 0) {  // lane 0 owns rows 0..7 / 16..23; lane 16 owns 8..15 / 24..31
        const int rbase = (lane >> 4) * 8;
#pragma unroll
        for (int r = 0; r < 8; ++r) {
            atomicAdd(&lds_dist[rbase + r],      dsum0[r]);
            atomicAdd(&lds_dist[rbase + r + 16], dsum1[r]);
        }
    }
    __syncthreads();

    // ---- Phase 4: scores[b, c] = -dist[b] ----
    if (t < B_SZ) out[t * C_SZ + c] = -lds_dist[t];
}

extern "C" void kernel_launch(void* const* d_in, const int* in_sizes, int n_in,
                              void* d_out, int out_size, void* d_ws, size_t ws_size,
                              hipStream_t stream) {
    const float* x      = (const float*)d_in[0];  // raw_features  [32, 768]
    const float* means  = (const float*)d_in[1];  // class_means   [256, 768]
    const float* diags  = (const float*)d_in[2];  // class_diags   [256, 768]
    const float* covinv = (const float*)d_in[3];  // class_cov_invs[256, 768, 768]
    float* out = (float*)d_out;                   // scores        [32, 256]

    fecam_score_kernel<<<dim3(C_SZ), dim3(256), 0, stream>>>(x, means, diags, covinv, out);
}